// TorchGradientBoostingClassifier_63282048139717
// MI455X (gfx1250) — compile-verified
//
#include <hip/hip_runtime.h>
#include <stdint.h>

// ---------------------------------------------------------------------------
// Gradient-boosted forest inference (FIL-style), CDNA5 / gfx1250.
//   B=50000 samples x F=100 features, T=2000 complete depth-6 trees,
//   10 classes, lr=0.1, output = log_softmax(decisions)  [B,10] f32.
//
// Block = 128 threads = 128 samples. x tile staged into LDS once with
// per-lane GLOBAL_LOAD_ASYNC_TO_LDS_B128 (ASYNCcnt). Tree tables stream
// through LDS in double-buffered 40-tree chunks moved by the Tensor Data
// Mover (TENSOR_LOAD_TO_LDS, one descriptor per table, issued by wave 0,
// synchronized with s_wait_tensorcnt + barrier). Each thread traverses all
// 2000 trees for its sample with 10 register class-accumulators, then does
// the log-softmax epilogue and writes d_out. No atomics, no workspace.
// ---------------------------------------------------------------------------

namespace {

constexpr int F_DIM  = 100;   // features per sample
constexpr int DEPTH  = 6;     // tree depth
constexpr int NI     = 63;    // internal nodes per tree
constexpr int NL     = 64;    // leaves per tree
constexpr int NCLS   = 10;    // classes
constexpr int S_TILE = 128;   // samples per block == threads per block
constexpr int TC     = 40;    // trees per LDS chunk (4 stages x 10 classes)
constexpr float LR   = 0.1f;

// LDS layout (bytes). Everything 16-byte aligned.
constexpr uint32_t X_BYTES   = S_TILE * F_DIM * 4;      // 51200
constexpr uint32_t FEAT_BUF  = TC * NI * 4;             // 10080
constexpr uint32_t LEAF_BUF  = TC * NL * 4;             // 10240
constexpr uint32_t LDS_X     = 0;
constexpr uint32_t LDS_FEAT  = X_BYTES;                 // 51200
constexpr uint32_t LDS_THR   = LDS_FEAT + 2 * FEAT_BUF; // 71360
constexpr uint32_t LDS_LEAF  = LDS_THR + 2 * FEAT_BUF;  // 91520
constexpr uint32_t LDS_TOTAL = LDS_LEAF + 2 * LEAF_BUF; // 112000

typedef uint32_t u32x4 __attribute__((ext_vector_type(4)));
typedef uint32_t u32x8 __attribute__((ext_vector_type(8)));

// ---- per-lane async global->LDS copy (ASYNCcnt) ---------------------------
__device__ __forceinline__ void async_b128(uint32_t lds_off, uint64_t gbase,
                                           uint32_t goff) {
  asm volatile("global_load_async_to_lds_b128 %0, %1, %2"
               :
               : "v"(lds_off), "v"(goff), "s"(gbase)
               : "memory");
}

__device__ __forceinline__ void wait_async0() {
  asm volatile("s_wait_asynccnt 0" ::: "memory");
}

// ---- Tensor Data Mover: 1-D DMA of `tile_e8` 8-byte elements --------------
// D# group0: count=1 | lds_addr | global_addr(57b) | type=2 ("image")
// D# group1: data_size=3 (8B) | tensor_dim0=rem_e8 (OOB->zero) | tensor_dim1=1
//            tile_dim0=tile_e8 | tile_dim1/2 unused | stride0=rem_e8
// Issued by one wave for the whole workgroup; tracked by TENSORcnt.
__device__ __forceinline__ void tdm_load_1d(uint32_t lds_off, uint64_t gaddr,
                                            uint32_t rem_e8, uint32_t tile_e8) {
  u32x4 g0;
  g0[0] = 0x1u;                                  // count=1, user descriptor
  g0[1] = lds_off;                               // lds_addr (bytes)
  g0[2] = (uint32_t)gaddr;                       // global_addr[31:0]
  g0[3] = (uint32_t)(gaddr >> 32) | (2u << 30);  // global_addr[56:32] | type=2
  u32x8 g1;
  g1[0] = 3u << 16;                              // wg_mask=0, data_size=8B
  g1[1] = (rem_e8 & 0xFFFFu) << 16;              // tensor_dim0[15:0]
  g1[2] = (rem_e8 >> 16) | (1u << 16);           // tensor_dim0[31:16]|dim1=1
  g1[3] = tile_e8 << 16;                         // dim1 hi=0 | tile_dim0
  g1[4] = 0u;                                    // tile_dim1=0, tile_dim2=0
  g1[5] = rem_e8;                                // tensor_dim0_stride[31:0]
  g1[6] = 0u;                                    // stride0 hi | stride1 lo
  g1[7] = 0u;                                    // stride1 hi
  asm volatile("tensor_load_to_lds %0, %1" : : "s"(g0), "s"(g1) : "memory");
}

// Issue TDM copies for tree chunk k into double-buffer half b (one wave).
__device__ __forceinline__ void tdm_tree_chunk(int k, int b, uint64_t featB,
                                               uint64_t thrB, uint64_t leafB) {
  const uint32_t fdst = LDS_FEAT + (uint32_t)b * FEAT_BUF;
  const uint32_t tdst = LDS_THR  + (uint32_t)b * FEAT_BUF;
  const uint32_t ldst = LDS_LEAF + (uint32_t)b * LEAF_BUF;
  const uint32_t fsrc = (uint32_t)k * FEAT_BUF;  // byte offsets into tables
  const uint32_t lsrc = (uint32_t)k * LEAF_BUF;
  tdm_load_1d(fdst, featB + fsrc, FEAT_BUF / 8, FEAT_BUF / 8);  // 1260 e8
  tdm_load_1d(tdst, thrB  + fsrc, FEAT_BUF / 8, FEAT_BUF / 8);  // 1260 e8
  tdm_load_1d(ldst, leafB + lsrc, LEAF_BUF / 8, LEAF_BUF / 8);  // 1280 e8
}

}  // namespace

__global__ __launch_bounds__(S_TILE) void fil_forest_kernel(
    const float* __restrict__ x, const float* __restrict__ thresholds,
    const float* __restrict__ leaf_values, const int* __restrict__ features,
    float* __restrict__ out, int B, int T) {
  extern __shared__ char smem[];
  const int tid    = threadIdx.x;
  const int sample = blockIdx.x * S_TILE + tid;
  const bool issuer = (tid < 32);  // wave 0 drives the TDM

  const uint64_t xB    = (uint64_t)(uintptr_t)x;
  const uint64_t thrB  = (uint64_t)(uintptr_t)thresholds;
  const uint64_t leafB = (uint64_t)(uintptr_t)leaf_values;
  const uint64_t featB = (uint64_t)(uintptr_t)features;

  // ---- prologue: stage x tile (per-lane async) + tree chunk 0 (TDM) -------
  if (issuer) tdm_tree_chunk(0, 0, featB, thrB, leafB);
  {
    const uint32_t xlimit = (uint32_t)B * (uint32_t)F_DIM * 4u - 16u;
    const uint32_t xbase  = (uint32_t)blockIdx.x * (uint32_t)(S_TILE * F_DIM * 4);
    for (int i = tid; i < (int)(X_BYTES / 16); i += S_TILE) {  // 3200 ops
      uint32_t g = xbase + (uint32_t)i * 16u;
      if (g > xlimit) g = xlimit;  // clamp tail block (garbage rows, unused)
      async_b128(LDS_X + (uint32_t)i * 16u, xB, g);
    }
  }
  wait_async0();
  __builtin_amdgcn_s_wait_tensorcnt(0);
  __syncthreads();

  const float* xrow = (const float*)(smem) + tid * F_DIM;

  float acc[NCLS];
#pragma unroll
  for (int c = 0; c < NCLS; ++c) acc[c] = 0.0f;

  const int nChunks = T / TC;  // 2000/40 = 50
  int b = 0;
  for (int k = 0; k < nChunks; ++k) {
    // DMA-prefetch next chunk into the other buffer (overlaps traversal)
    if (issuer && (k + 1 < nChunks))
      tdm_tree_chunk(k + 1, b ^ 1, featB, thrB, leafB);

    const int*   cf = (const int*)  (smem + LDS_FEAT + (uint32_t)b * FEAT_BUF);
    const float* ct = (const float*)(smem + LDS_THR  + (uint32_t)b * FEAT_BUF);
    const float* cl = (const float*)(smem + LDS_LEAF + (uint32_t)b * LEAF_BUF);

    // chunk = 4 stages x 10 classes; class index is compile-time so the 10
    // accumulators stay in registers and 10 traversal chains run in parallel.
    for (int s = 0; s < TC / NCLS; ++s) {
      const int base = s * NCLS;
#pragma unroll
      for (int c = 0; c < NCLS; ++c) {
        const int tl = base + c;       // tree within chunk
        const int nb = tl * NI;
        int idx = 0;
#pragma unroll
        for (int l = 0; l < DEPTH; ++l) {
          const int   f  = cf[nb + idx];   // ds_load_b32 gather
          const float tv = ct[nb + idx];   // ds_load_b32 gather
          const float xv = xrow[f];        // ds_load_b32 gather
          idx = 2 * idx + 1 + ((xv > tv) ? 1 : 0);
        }
        acc[c] += cl[tl * NL + (idx - NI)];
      }
    }

    __builtin_amdgcn_s_wait_tensorcnt(0);  // prefetched chunk resident
    __syncthreads();  // everyone done reading buffer b before it's reused
    b ^= 1;
  }

  // ---- epilogue: decisions = lr * stage-sums; log_softmax ------------------
  float d[NCLS];
  float m = -3.402823466e38f;
#pragma unroll
  for (int c = 0; c < NCLS; ++c) {
    d[c] = LR * acc[c];
    m = fmaxf(m, d[c]);
  }
  float ssum = 0.0f;
#pragma unroll
  for (int c = 0; c < NCLS; ++c) ssum += expf(d[c] - m);
  const float lse = m + logf(ssum);

  if (sample < B) {
#pragma unroll
    for (int c = 0; c < NCLS; ++c) out[sample * NCLS + c] = d[c] - lse;
  }
}

extern "C" void kernel_launch(void* const* d_in, const int* in_sizes, int n_in,
                              void* d_out, int out_size, void* d_ws,
                              size_t ws_size, hipStream_t stream) {
  (void)n_in; (void)out_size; (void)d_ws; (void)ws_size;
  const float* x          = (const float*)d_in[0];  // [B, 100]
  const float* thresholds = (const float*)d_in[1];  // [T, 63]
  const float* leaf_vals  = (const float*)d_in[2];  // [T, 64]
  const int*   features   = (const int*)d_in[3];    // [T, 63]
  // d_in[4] = classes (identity arange) -- unused: we return log-probs.

  const int B = in_sizes[0] / F_DIM;  // 50000
  const int T = in_sizes[1] / NI;     // 2000
  float* out = (float*)d_out;

  // Dynamic LDS is 112 KB (> legacy 64 KB default) -- raise the cap.
  (void)hipFuncSetAttribute((const void*)fil_forest_kernel,
                            hipFuncAttributeMaxDynamicSharedMemorySize,
                            (int)LDS_TOTAL);

  const int grid = (B + S_TILE - 1) / S_TILE;  // 391
  fil_forest_kernel<<<grid, S_TILE, LDS_TOTAL, stream>>>(
      x, thresholds, leaf_vals, features, out, B, T);
}